// SSDBBox_39633958207881
// MI455X (gfx1250) — compile-verified
//
#include <hip/hip_runtime.h>
#include <hip/hip_bf16.h>

// ---------------------------------------------------------------------------
// SSD bbox post-processing for gfx1250 (MI455X).
// Pipeline: softmax(81) -> candidate compaction (>0.02) -> exact radix top-1000
//           -> bbox decode -> sequential NMS -> sort -> top-100 outputs.
// CDNA5-specific paths: TDM tensor_load_to_lds (strided 2D tile -> LDS),
// s_wait_tensorcnt, wave32 sizing.
// ---------------------------------------------------------------------------

typedef unsigned int       v4u __attribute__((ext_vector_type(4)));
typedef int                v4i __attribute__((ext_vector_type(4)));
typedef int                v8i __attribute__((ext_vector_type(8)));
typedef unsigned long long u64;
typedef unsigned int       u32;

#define HW_        80
#define NS_        6400      // 80*80 spatial positions
#define NA_        6         // anchors per position
#define C81_       81
#define C80_       80
#define NPRE_      1000
#define SCORE_THR_ 0.02f
#define IOU_THR_   0.45f
#define MAXR_      4.135166556742356f   // |log(0.016)|
#define MAXS_      320.0f
#define TILE_S_    128       // spatial columns per block

#if defined(__has_builtin)
#if __has_builtin(__builtin_amdgcn_tensor_load_to_lds)
#define HAVE_TDM 1
#endif
#endif

__global__ void init_kernel(u32* cnt) { *cnt = 0u; }

// ---------------------------------------------------------------------------
// K1: softmax over 81 classes per (spatial, anchor) row; emit candidate keys
// (score_bits<<32 | ~flat_index) for scores > 0.02.
// Grid: (50 spatial tiles, 6 anchors), 128 threads (4 wave32s).
// The 81x128 tile (row stride 6400 floats in global) is DMA'd to LDS by the
// Tensor Data Mover, issued by wave 0 only.
// ---------------------------------------------------------------------------
__global__ void score_kernel(const float* __restrict__ cls,
                             u64* __restrict__ cand, u32* __restrict__ cnt,
                             u32 cap) {
  __shared__ float tile[C81_ * TILE_S_];   // 41.5 KB of the 320 KB WGP LDS
  const int t  = threadIdx.x;
  const int s0 = blockIdx.x * TILE_S_;
  const int a  = blockIdx.y;
  const float* base = cls + (size_t)a * C81_ * NS_ + s0;

#if HAVE_TDM
  if (t < 32) {  // wave 0 issues the TDM descriptor (EXEC ignored by TDM)
    const u64 ga   = (u64)(size_t)base;
    const u32 ldsa = (u32)(size_t)&tile[0];
    v4u g0;
    g0.x = 1u;                               // count=1, user descriptor
    g0.y = ldsa;                             // lds_addr
    g0.z = (u32)(ga & 0xFFFFFFFFu);          // global_addr[31:0]
    g0.w = (u32)((ga >> 32) & 0x01FFFFFFu) | (2u << 30); // ga[56:32] | type=2
    v8i g1;
    g1[0] = (int)(2u << 16);                 // data_size=2 (4 bytes)
    g1[1] = (int)((NS_ & 0xFFFFu) << 16);    // tensor_dim0 lo16 @ bits 63:48
    g1[2] = (int)(((u32)NS_ >> 16) & 0xFFFFu) | (int)(((u32)C81_ & 0xFFFFu) << 16);
    g1[3] = (int)(((u32)C81_ >> 16) & 0xFFFFu) | (int)((u32)TILE_S_ << 16); // tile_dim0
    g1[4] = (int)C81_;                       // tile_dim1=81, tile_dim2=0
    g1[5] = (int)NS_;                        // tensor_dim0_stride lo32
    g1[6] = 0;                               // stride hi / dim1_stride lo
    g1[7] = 0;
    v4i g2 = {0, 0, 0, 0};                   // 2D tensor: groups 2/3 unused
    v4i g3 = {0, 0, 0, 0};
    v8i g4 = {0, 0, 0, 0, 0, 0, 0, 0};       // extra group (unused for 2D)
    __builtin_amdgcn_tensor_load_to_lds(g0, g1, g2, g3, g4, 0);
#if __has_builtin(__builtin_amdgcn_s_wait_tensorcnt)
    __builtin_amdgcn_s_wait_tensorcnt(0);
#else
    asm volatile("s_wait_tensorcnt 0" ::: "memory");
#endif
  }
  __syncthreads();
#else
  // Fallback: coalesced cooperative staging (81 rows, 128 cols).
  for (int k = 0; k < C81_; ++k) tile[k * TILE_S_ + t] = base[(size_t)k * NS_ + t];
  __syncthreads();
#endif

  // Per-thread softmax over its column (bank-conflict-free: stride 128 dwords).
  float m = -1e30f;
  for (int k = 0; k < C81_; ++k) m = fmaxf(m, tile[k * TILE_S_ + t]);
  float ssum = 0.0f;
  for (int k = 0; k < C81_; ++k) ssum += __expf(tile[k * TILE_S_ + t] - m);
  const float inv = 1.0f / ssum;
  const int row = (s0 + t) * NA_ + a;

  for (int k = 0; k < C80_; ++k) {
    const float sc = __expf(tile[k * TILE_S_ + t] - m) * inv;
    if (sc > SCORE_THR_) {
      const u32 pos = atomicAdd(cnt, 1u);
      if (pos < cap) {
        const u32 idx = (u32)(row * C80_ + k);
        cand[pos] = ((u64)__float_as_uint(sc) << 32) | (u32)(~idx);
      }
    }
  }
}

// ---------------------------------------------------------------------------
// K2: exact 64-bit radix select of the 1000th-largest key (keys are unique).
// Single block, 1024 threads, 8 byte-passes over the candidate list.
// ---------------------------------------------------------------------------
__global__ void select_kernel(const u64* __restrict__ cand,
                              const u32* __restrict__ cnt, u32 cap,
                              u64* __restrict__ pivot_out) {
  __shared__ u32 hist[256];
  __shared__ u64 sh_prefix;
  __shared__ u32 sh_rem;
  __shared__ u32 sh_n;
  const int t = threadIdx.x;
  if (t == 0) {
    u32 n = *cnt; if (n > cap) n = cap;
    sh_n = n; sh_rem = NPRE_; sh_prefix = 0ULL;
  }
  __syncthreads();
  const u32 n = sh_n;
  if (n <= NPRE_) { if (t == 0) *pivot_out = 0ULL; return; }

  for (int b = 7; b >= 0; --b) {
    if (t < 256) hist[t] = 0u;
    __syncthreads();
    const u64 pref = sh_prefix;
    for (u32 i = (u32)t; i < n; i += 1024u) {
      const u64 key = cand[i];
      const bool match = (b == 7) || ((key >> (u32)((b + 1) * 8)) == pref);
      if (match) atomicAdd(&hist[(u32)(key >> (u32)(b * 8)) & 0xFFu], 1u);
    }
    __syncthreads();
    if (t == 0) {
      u32 rem = sh_rem, cum = 0u; int chosen = 0;
      for (int bin = 255; bin >= 0; --bin) {
        const u32 c = hist[bin];
        if (cum + c >= rem) { chosen = bin; sh_rem = rem - cum; break; }
        cum += c;
      }
      sh_prefix = (sh_prefix << 8) | (u32)chosen;
    }
    __syncthreads();
  }
  if (t == 0) *pivot_out = sh_prefix;
}

// ---------------------------------------------------------------------------
// K3: gather keys >= pivot (exactly min(n,1000), keys unique), pad to 1024,
// bitonic sort descending in LDS. Result reproduces jax top_k order exactly
// (score desc, index asc via the ~index low word).
// ---------------------------------------------------------------------------
__global__ void compact_sort_kernel(const u64* __restrict__ cand,
                                    const u32* __restrict__ cnt, u32 cap,
                                    const u64* __restrict__ pivot_in,
                                    u64* __restrict__ sorted_out) {
  __shared__ u64 sel[1024];
  __shared__ u32 scount;
  const int t = threadIdx.x;
  if (t == 0) scount = 0u;
  sel[t] = 0ULL;
  __syncthreads();
  u32 n = *cnt; if (n > cap) n = cap;
  const u64 pivot = *pivot_in;
  for (u32 i = (u32)t; i < n; i += 1024u) {
    const u64 key = cand[i];
    if (key != 0ULL && key >= pivot) {
      const u32 pos = atomicAdd(&scount, 1u);
      if (pos < 1024u) sel[pos] = key;
    }
  }
  __syncthreads();
  for (int k = 2; k <= 1024; k <<= 1) {
    for (int j = k >> 1; j > 0; j >>= 1) {
      const int ij = t ^ j;
      if (ij > t) {
        const u64 a = sel[t], c = sel[ij];
        const bool up = ((t & k) == 0);
        if (up ? (a < c) : (a > c)) { sel[t] = c; sel[ij] = a; }
      }
      __syncthreads();
    }
  }
  sorted_out[t] = sel[t];
}

// ---------------------------------------------------------------------------
// K4: decode 1000 candidates: priors + delta2bbox, max_coord reduction,
// class-offset boxes for NMS.
// ---------------------------------------------------------------------------
__global__ void decode_kernel(const u64* __restrict__ sorted,
                              const float* __restrict__ bbp,
                              const float* __restrict__ banch,
                              const int* __restrict__ stride_p,
                              float* __restrict__ bbox, float* __restrict__ scr,
                              int* __restrict__ lab, int* __restrict__ valid,
                              float* __restrict__ offb, float* __restrict__ maxc) {
  __shared__ float red[1024];
  const int t = threadIdx.x;
  const float stride = (float)stride_p[0];
  float mymax = -1e30f;
  float bx0 = 0.f, bx1 = 0.f, bx2 = 0.f, bx3 = 0.f;
  int lb = 0;
  if (t < NPRE_) {
    const u64 key = sorted[t];
    const float sc = __uint_as_float((u32)(key >> 32));
    const u32 idx = (key == 0ULL) ? 0u : ~((u32)key);
    lb = (int)(idx % (u32)C80_);
    const u32 row = idx / (u32)C80_;
    const u32 s = row / (u32)NA_, a = row % (u32)NA_;
    const float shx = (float)(s % (u32)HW_) * stride;
    const float shy = (float)(s / (u32)HW_) * stride;
    const float p0 = banch[a * 4 + 0] + shx;
    const float p1 = banch[a * 4 + 1] + shy;
    const float p2 = banch[a * 4 + 2] + shx;
    const float p3 = banch[a * 4 + 3] + shy;
    const float d0 = bbp[(size_t)(a * 4 + 0) * NS_ + s] * 0.1f;
    const float d1 = bbp[(size_t)(a * 4 + 1) * NS_ + s] * 0.1f;
    float d2 = bbp[(size_t)(a * 4 + 2) * NS_ + s] * 0.2f;
    float d3 = bbp[(size_t)(a * 4 + 3) * NS_ + s] * 0.2f;
    d2 = fminf(fmaxf(d2, -MAXR_), MAXR_);
    d3 = fminf(fmaxf(d3, -MAXR_), MAXR_);
    const float px = (p0 + p2) * 0.5f, py = (p1 + p3) * 0.5f;
    const float pw = p2 - p0, ph = p3 - p1;
    const float gx = px + pw * d0, gy = py + ph * d1;
    const float gw = pw * __expf(d2), gh = ph * __expf(d3);
    bx0 = fminf(fmaxf(gx - gw * 0.5f, 0.f), MAXS_);
    bx1 = fminf(fmaxf(gy - gh * 0.5f, 0.f), MAXS_);
    bx2 = fminf(fmaxf(gx + gw * 0.5f, 0.f), MAXS_);
    bx3 = fminf(fmaxf(gy + gh * 0.5f, 0.f), MAXS_);
    bbox[t * 4 + 0] = bx0; bbox[t * 4 + 1] = bx1;
    bbox[t * 4 + 2] = bx2; bbox[t * 4 + 3] = bx3;
    mymax = fmaxf(fmaxf(bx0, bx1), fmaxf(bx2, bx3));
    scr[t] = sc; lab[t] = lb; valid[t] = (sc > SCORE_THR_) ? 1 : 0;
  }
  red[t] = mymax;
  __syncthreads();
  for (int sft = 512; sft > 0; sft >>= 1) {
    if (t < sft) red[t] = fmaxf(red[t], red[t + sft]);
    __syncthreads();
  }
  const float mc = red[0];
  if (t == 0) *maxc = mc;
  if (t < NPRE_) {
    const float off = (float)lb * (mc + 1.0f);
    offb[t * 4 + 0] = bx0 + off; offb[t * 4 + 1] = bx1 + off;
    offb[t * 4 + 2] = bx2 + off; offb[t * 4 + 3] = bx3 + off;
  }
}

// ---------------------------------------------------------------------------
// K5: sequential greedy NMS (1000 steps), boxes/areas/keep in LDS.
// ---------------------------------------------------------------------------
__global__ void nms_kernel(const float* __restrict__ offb,
                           const int* __restrict__ valid,
                           u32* __restrict__ keep_out) {
  __shared__ float b[NPRE_ * 4];
  __shared__ float area[NPRE_];
  __shared__ u32 keep[NPRE_];
  const int t = threadIdx.x;
  for (int i = t; i < NPRE_; i += 1024) {
    const float x1 = offb[i * 4 + 0], y1 = offb[i * 4 + 1];
    const float x2 = offb[i * 4 + 2], y2 = offb[i * 4 + 3];
    b[i * 4 + 0] = x1; b[i * 4 + 1] = y1; b[i * 4 + 2] = x2; b[i * 4 + 3] = y2;
    area[i] = (x2 - x1) * (y2 - y1);
    keep[i] = (u32)valid[i];
  }
  __syncthreads();
  for (int i = 0; i < NPRE_ - 1; ++i) {
    if (t > i && t < NPRE_ && keep[i] && keep[t]) {
      const float lx = fmaxf(b[i * 4 + 0], b[t * 4 + 0]);
      const float ly = fmaxf(b[i * 4 + 1], b[t * 4 + 1]);
      const float rx = fminf(b[i * 4 + 2], b[t * 4 + 2]);
      const float ry = fminf(b[i * 4 + 3], b[t * 4 + 3]);
      const float w = fmaxf(rx - lx, 0.f), h = fmaxf(ry - ly, 0.f);
      const float inter = w * h;
      const float iou = inter / (area[i] + area[t] - inter);
      if (iou > IOU_THR_) keep[t] = 0u;
    }
    __syncthreads();
  }
  for (int i = t; i < NPRE_; i += 1024) keep_out[i] = keep[i];
}

// ---------------------------------------------------------------------------
// K6: key = keep ? score : -1, stable descending argsort (bitonic on
// order-preserving u64 keys), emit top-100 boxes/scores/labels.
// ---------------------------------------------------------------------------
__global__ void output_kernel(const float* __restrict__ bbox,
                              const float* __restrict__ scr,
                              const int* __restrict__ lab,
                              const u32* __restrict__ keep,
                              float* __restrict__ out) {
  __shared__ u64 key[1024];
  const int t = threadIdx.x;
  u64 kv = 0ULL;
  if (t < NPRE_) {
    const float v = keep[t] ? scr[t] : -1.0f;
    u32 u = __float_as_uint(v);
    u = (u & 0x80000000u) ? ~u : (u | 0x80000000u);
    kv = ((u64)u << 32) | (u32)(~(u32)t);
  }
  key[t] = kv;
  __syncthreads();
  for (int k = 2; k <= 1024; k <<= 1) {
    for (int j = k >> 1; j > 0; j >>= 1) {
      const int ij = t ^ j;
      if (ij > t) {
        const u64 a = key[t], c = key[ij];
        const bool up = ((t & k) == 0);
        if (up ? (a < c) : (a > c)) { key[t] = c; key[ij] = a; }
      }
      __syncthreads();
    }
  }
  if (t < 100) {
    const u64 k2 = key[t];
    const u32 orig = ~((u32)k2);   // top-100 are always real entries
    const u32 kp = keep[orig];
    out[t * 5 + 0] = kp ? bbox[orig * 4 + 0] : 0.f;
    out[t * 5 + 1] = kp ? bbox[orig * 4 + 1] : 0.f;
    out[t * 5 + 2] = kp ? bbox[orig * 4 + 2] : 0.f;
    out[t * 5 + 3] = kp ? bbox[orig * 4 + 3] : 0.f;
    out[t * 5 + 4] = kp ? scr[orig] : 0.f;
    out[500 + t]   = kp ? (float)lab[orig] : -1.0f;
  }
}

// ---------------------------------------------------------------------------
extern "C" void kernel_launch(void* const* d_in, const int* in_sizes, int n_in,
                              void* d_out, int out_size, void* d_ws, size_t ws_size,
                              hipStream_t stream) {
  const float* cls      = (const float*)d_in[0];  // (1, 486, 80, 80)
  const float* bbp      = (const float*)d_in[1];  // (1, 24, 80, 80)
  const float* banch    = (const float*)d_in[2];  // (6, 4)
  const int*   stride_p = (const int*)d_in[3];    // scalar 4
  float* out = (float*)d_out;                     // 100*5 + 100 = 600

  char* ws = (char*)d_ws;
  u32* cnt    = (u32*)(ws + 0);
  u64* pivot  = (u64*)(ws + 8);
  u64* sorted = (u64*)(ws + 16);                  // 1024 * 8 = 8192 B
  float* bbox = (float*)(ws + 8208);              // 1000*4 f
  float* scr  = (float*)(ws + 24208);             // 1000 f
  int*   lab  = (int*)  (ws + 28208);             // 1000 i
  int*   vld  = (int*)  (ws + 32208);             // 1000 i
  float* offb = (float*)(ws + 36208);             // 1000*4 f
  float* maxc = (float*)(ws + 52208);             // 1 f
  u32*   keep = (u32*)  (ws + 52224);             // 1000 u32
  u64*   cand = (u64*)  (ws + 57344);             // candidate keys
  size_t cap64 = (ws_size > 57344) ? (ws_size - 57344) / 8 : 0;
  u32 cap = (u32)((cap64 > (size_t)(1u << 20)) ? (1u << 20) : cap64);

  init_kernel<<<1, 1, 0, stream>>>(cnt);
  score_kernel<<<dim3(NS_ / TILE_S_, NA_), TILE_S_, 0, stream>>>(cls, cand, cnt, cap);
  select_kernel<<<1, 1024, 0, stream>>>(cand, cnt, cap, pivot);
  compact_sort_kernel<<<1, 1024, 0, stream>>>(cand, cnt, cap, pivot, sorted);
  decode_kernel<<<1, 1024, 0, stream>>>(sorted, bbp, banch, stride_p,
                                        bbox, scr, lab, vld, offb, maxc);
  nms_kernel<<<1, 1024, 0, stream>>>(offb, vld, keep);
  output_kernel<<<1, 1024, 0, stream>>>(bbox, scr, lab, keep, out);
}